// TreeGRU_33913061769570
// MI455X (gfx1250) — compile-verified
//
#include <hip/hip_runtime.h>

// h[n, o] = sum_i x[n,i] * W[o,i] + b[o]
// N = 4,194,304 rows, IN = HID = 64. f32 throughout (matches reference).
//
// Memory-bound problem (~2.1 GB @ 23.3 TB/s => ~92us floor). Strategy:
//  - W (16KB) -> LDS once per block, B-fragments hoisted to registers.
//  - x streamed via CDNA5 async global->LDS (ASYNCcnt), double buffered.
//  - f32 WMMA 16x16x4, 4 N-tiles x 16 K-steps per 16-row stripe per wave.
//  - LDS rows padded to 68 floats => conflict-free ds_load_b64 fragments.

typedef float v2f __attribute__((ext_vector_type(2)));
typedef float v8f __attribute__((ext_vector_type(8)));

#define LSTR        68              // padded LDS row stride (floats)
#define CHUNK_ROWS  128             // rows per double-buffered chunk (= 8 waves * 16)
#define NCHUNK      4               // chunks per block
#define BLOCK_ROWS  (CHUNK_ROWS * NCHUNK)   // 512 rows per block

__device__ __forceinline__ unsigned lds_off(const void* p) {
  // gfx1250 flat map: low 32 bits of a generic pointer into LDS == LDS byte offset
  return (unsigned)(unsigned long long)p;
}

__device__ __forceinline__ void async_ld_b128(unsigned lds, const float* g) {
  asm volatile("global_load_async_to_lds_b128 %0, %1, off"
               :: "v"(lds), "v"(g) : "memory");
}

__device__ __forceinline__ void issue_chunk(const float* __restrict__ x, size_t row0,
                                            float* sx, int tid) {
  // 128 rows x 64 cols = 2048 float4s; 256 threads x 8 b128 asyncs, coalesced
#pragma unroll
  for (int q = 0; q < 8; ++q) {
    int f  = tid + 256 * q;     // float4 index
    int r  = f >> 4;            // row in chunk
    int c4 = f & 15;            // float4 column
    async_ld_b128(lds_off(&sx[r * LSTR + c4 * 4]),
                  x + (row0 + (size_t)r) * 64 + c4 * 4);
  }
}

extern "C" __global__ void __launch_bounds__(256)
leaf_linear_wmma(const float* __restrict__ x, const float* __restrict__ W,
                 const float* __restrict__ bias, float* __restrict__ out) {
  __shared__ __align__(16) float sW[64 * LSTR];               // 17.0 KB
  __shared__ __align__(16) float sX[2][CHUNK_ROWS * LSTR];    // 2 x 34.0 KB

  const int tid  = threadIdx.x;
  const int lane = tid & 31;
  const int wv   = tid >> 5;        // wave in block: 0..7
  const int hh   = lane >> 4;       // half-wave: 0/1 (selects K pair / M+8)
  const int j    = lane & 15;       // lane within half: M (A), N (B/C/D)

  const size_t blockRow0 = (size_t)blockIdx.x * BLOCK_ROWS;

  // ---- issue W -> LDS (4 asyncs/thread), then chunk0 + chunk1 (8 each) ----
#pragma unroll
  for (int q = 0; q < 4; ++q) {
    int f = tid + 256 * q;          // float4 index in 64x64 W
    int r = f >> 4, c4 = f & 15;
    async_ld_b128(lds_off(&sW[r * LSTR + c4 * 4]), W + r * 64 + c4 * 4);
  }
  issue_chunk(x, blockRow0 + 0 * (size_t)CHUNK_ROWS, sX[0], tid);
  issue_chunk(x, blockRow0 + 1 * (size_t)CHUNK_ROWS, sX[1], tid);

  // outstanding: 4(W) + 8(c0) + 8(c1); wait until only chunk1 remains
  asm volatile("s_wait_asynccnt 8" ::: "memory");
  __syncthreads();

  // ---- hoist B fragments (Wt tile, all 4 N-tiles x 16 K-steps) to registers ----
  // B[kk, n] = W[n, kk]; lane j -> N = 16t + j; VGPR pair holds K = 4k+2hh, +1
  v2f bf[4][16];
#pragma unroll
  for (int t = 0; t < 4; ++t) {
    const int n = 16 * t + j;
#pragma unroll
    for (int k = 0; k < 16; ++k)
      bf[t][k] = *(const v2f*)&sW[n * LSTR + 4 * k + 2 * hh];
  }
  float bb[4];
#pragma unroll
  for (int t = 0; t < 4; ++t) bb[t] = bias[16 * t + j];

  const int m0 = wv * 16;           // this wave's row stripe within the chunk

#pragma unroll
  for (int c = 0; c < NCHUNK; ++c) {
    const float* sx = sX[c & 1];

    v8f acc[4];
#pragma unroll
    for (int t = 0; t < 4; ++t)
#pragma unroll
      for (int e = 0; e < 8; ++e) acc[t][e] = 0.0f;

    // inner loop: 16 ds_load_b64 (A) + 64 v_wmma_f32_16x16x4_f32
#pragma unroll
    for (int k = 0; k < 16; ++k) {
      v2f a = *(const v2f*)&sx[(m0 + j) * LSTR + 4 * k + 2 * hh];
#pragma unroll
      for (int t = 0; t < 4; ++t)
        acc[t] = __builtin_amdgcn_wmma_f32_16x16x4_f32(
            false, a, false, bf[t][k], (short)0, acc[t], false, false);
    }

    // D layout: VGPR r -> row m0 + r + 8*hh, lane j -> col 16t + j
    const size_t rowBase = blockRow0 + (size_t)c * CHUNK_ROWS + m0 + hh * 8;
#pragma unroll
    for (int t = 0; t < 4; ++t)
#pragma unroll
      for (int r = 0; r < 8; ++r)
        __builtin_nontemporal_store(acc[t][r] + bb[t],
                                    &out[(rowBase + r) * 64 + 16 * t + j]);

    __syncthreads();                // everyone finished reading sX[c&1]
    if (c + 2 < NCHUNK)
      issue_chunk(x, blockRow0 + (size_t)(c + 2) * CHUNK_ROWS, sX[c & 1], tid);
    if (c + 1 < NCHUNK) {
      if (c + 2 < NCHUNK) asm volatile("s_wait_asynccnt 8" ::: "memory");
      else                asm volatile("s_wait_asynccnt 0" ::: "memory");
      __syncthreads();              // chunk c+1 fully resident
    }
  }
}

extern "C" void kernel_launch(void* const* d_in, const int* in_sizes, int n_in,
                              void* d_out, int out_size, void* d_ws, size_t ws_size,
                              hipStream_t stream) {
  const float* x    = (const float*)d_in[0];
  const float* W    = (const float*)d_in[1];
  const float* bias = (const float*)d_in[2];
  float* out        = (float*)d_out;

  const long long n = (long long)in_sizes[0] / 64;   // rows (4,194,304; divisible by 512)
  const int blocks  = (int)(n / BLOCK_ROWS);
  leaf_linear_wmma<<<blocks, 256, 0, stream>>>(x, W, bias, out);
}